// CosScorer_72859825209884
// MI455X (gfx1250) — compile-verified
//
#include <hip/hip_runtime.h>
#include <math.h>

typedef float v2f __attribute__((ext_vector_type(2)));
typedef float v8f __attribute__((ext_vector_type(8)));

#define COS_EPS 1e-8f

// Geometry (fixed by the reference problem)
#define BATCH 8
#define TDIM  4096
#define SDIM  256
#define DDIM  512

// ---------------------------------------------------------------------------
// Row inverse-norm: one wave (32 lanes) per row of 512 f32.
// block = 256 threads = 8 waves = 8 rows per block.
// inv[row] = 1 / max(||x_row||_2, eps)
// ---------------------------------------------------------------------------
__global__ __launch_bounds__(256) void cos_rownorm_kernel(
    const float* __restrict__ x, float* __restrict__ inv, int nrows)
{
    const int lane = threadIdx.x & 31;
    const int wv   = threadIdx.x >> 5;
    const int row  = blockIdx.x * 8 + wv;
    if (row >= nrows) return;

    const float* p = x + (size_t)row * DDIM;
    float s = 0.0f;
#pragma unroll
    for (int i = 0; i < DDIM / 32; ++i) {
        float v = p[lane + 32 * i];
        s += v * v;
    }
    // wave32 butterfly reduction
#pragma unroll
    for (int off = 16; off > 0; off >>= 1)
        s += __shfl_xor(s, off, 32);

    if (lane == 0)
        inv[row] = 1.0f / fmaxf(sqrtf(s), COS_EPS);
}

// ---------------------------------------------------------------------------
// Batched GEMM with full-precision f32 WMMA (v_wmma_f32_16x16x4_f32).
// Workgroup = 256 threads (8 waves), waves arranged 4(M) x 2(N).
// Block tile = 128(M) x 128(N); per-wave tile = 32(M) x 64(N):
//   2 A-fragments x 4 B-fragments -> 8 v8f accumulators,
//   6 loads per 8 WMMAs (0.75 loads/WMMA) to keep VMEM issue off the
//   matrix-issue critical path.
// A = X[b] (T x D row-major), B = Y[b] (S x D row-major) -> scores = A B^T.
// Both operands are K-contiguous; 32-bit 16x4 fragment layout: lane holds
// row/col (lane&15), K = k + 2*(lane>>4) + {0,1} -> aligned float2 loads.
// Normalization applied at store: out = dot * rnx[row] * rny[col].
// ---------------------------------------------------------------------------
__global__ __launch_bounds__(256, 2) void cos_wmma_gemm_kernel(
    const float* __restrict__ X, const float* __restrict__ Y,
    const float* __restrict__ rnx, const float* __restrict__ rny,
    float* __restrict__ out)
{
    const int lane = threadIdx.x & 31;
    const int wv   = threadIdx.x >> 5;
    const int wv_m = wv & 3;          // 0..3
    const int wv_n = wv >> 2;         // 0..1
    const int b    = blockIdx.z;
    const int mBase = blockIdx.x * 128 + wv_m * 32;
    const int nBase = blockIdx.y * 128 + wv_n * 64;

    const int lh = lane >> 4;   // 0 or 1 (selects K pair)
    const int ll = lane & 15;   // row/col within 16-tile

    const float* A  = X + (size_t)b * TDIM * DDIM;
    const float* Bm = Y + (size_t)b * SDIM * DDIM;

    // Fragment base pointers (float2 = one lane's 2 consecutive K values)
    const v2f* ap0 = (const v2f*)(A  + (size_t)(mBase +  0 + ll) * DDIM) + lh;
    const v2f* ap1 = (const v2f*)(A  + (size_t)(mBase + 16 + ll) * DDIM) + lh;
    const v2f* bp0 = (const v2f*)(Bm + (size_t)(nBase +  0 + ll) * DDIM) + lh;
    const v2f* bp1 = (const v2f*)(Bm + (size_t)(nBase + 16 + ll) * DDIM) + lh;
    const v2f* bp2 = (const v2f*)(Bm + (size_t)(nBase + 32 + ll) * DDIM) + lh;
    const v2f* bp3 = (const v2f*)(Bm + (size_t)(nBase + 48 + ll) * DDIM) + lh;

    v8f acc00 = {}, acc01 = {}, acc02 = {}, acc03 = {};
    v8f acc10 = {}, acc11 = {}, acc12 = {}, acc13 = {};

#pragma unroll 2
    for (int k = 0; k < DDIM; k += 4) {
        const int ki = k >> 1;      // float2 index; lane offset already applied
        v2f a0 = ap0[ki];
        v2f a1 = ap1[ki];
        v2f b0 = bp0[ki];
        v2f b1 = bp1[ki];
        v2f b2 = bp2[ki];
        v2f b3 = bp3[ki];
        // 8-arg form: (neg_a, A, neg_b, B, c_mod, C, reuse_a, reuse_b)
        acc00 = __builtin_amdgcn_wmma_f32_16x16x4_f32(false, a0, false, b0,
                                                      (short)0, acc00, false, false);
        acc01 = __builtin_amdgcn_wmma_f32_16x16x4_f32(false, a0, false, b1,
                                                      (short)0, acc01, false, false);
        acc02 = __builtin_amdgcn_wmma_f32_16x16x4_f32(false, a0, false, b2,
                                                      (short)0, acc02, false, false);
        acc03 = __builtin_amdgcn_wmma_f32_16x16x4_f32(false, a0, false, b3,
                                                      (short)0, acc03, false, false);
        acc10 = __builtin_amdgcn_wmma_f32_16x16x4_f32(false, a1, false, b0,
                                                      (short)0, acc10, false, false);
        acc11 = __builtin_amdgcn_wmma_f32_16x16x4_f32(false, a1, false, b1,
                                                      (short)0, acc11, false, false);
        acc12 = __builtin_amdgcn_wmma_f32_16x16x4_f32(false, a1, false, b2,
                                                      (short)0, acc12, false, false);
        acc13 = __builtin_amdgcn_wmma_f32_16x16x4_f32(false, a1, false, b3,
                                                      (short)0, acc13, false, false);
    }

    // Per-lane column inverse norms (one per N-subtile)
    const float ry0 = rny[b * SDIM + nBase +  0 + ll];
    const float ry1 = rny[b * SDIM + nBase + 16 + ll];
    const float ry2 = rny[b * SDIM + nBase + 32 + ll];
    const float ry3 = rny[b * SDIM + nBase + 48 + ll];

    float* obase = out + (size_t)b * TDIM * SDIM;

    // C/D layout: VGPR i, lanes 0-15 -> M = i, N = lane; lanes 16-31 -> M = i+8
#pragma unroll
    for (int i = 0; i < 8; ++i) {
        {   // M stripe 0: rows mBase + i + 8*lh
            const int row = mBase + i + 8 * lh;
            const float rx = rnx[b * TDIM + row];
            float* orow = obase + (size_t)row * SDIM + nBase + ll;
            orow[0]  = acc00[i] * rx * ry0;
            orow[16] = acc01[i] * rx * ry1;
            orow[32] = acc02[i] * rx * ry2;
            orow[48] = acc03[i] * rx * ry3;
        }
        {   // M stripe 1: rows mBase + 16 + i + 8*lh
            const int row = mBase + 16 + i + 8 * lh;
            const float rx = rnx[b * TDIM + row];
            float* orow = obase + (size_t)row * SDIM + nBase + ll;
            orow[0]  = acc10[i] * rx * ry0;
            orow[16] = acc11[i] * rx * ry1;
            orow[32] = acc12[i] * rx * ry2;
            orow[48] = acc13[i] * rx * ry3;
        }
    }
}

extern "C" void kernel_launch(void* const* d_in, const int* in_sizes, int n_in,
                              void* d_out, int out_size, void* d_ws, size_t ws_size,
                              hipStream_t stream) {
    const float* xs  = (const float*)d_in[0];  // (8, 4096, 512) f32
    const float* spk = (const float*)d_in[1];  // (8, 256, 512) f32
    float* out = (float*)d_out;                // (8, 4096, 256) f32

    float* rnx = (float*)d_ws;                 // 8*4096 inverse norms
    float* rny = rnx + BATCH * TDIM;           // 8*256 inverse norms

    const int xrows = BATCH * TDIM;            // 32768
    const int yrows = BATCH * SDIM;            // 2048

    cos_rownorm_kernel<<<xrows / 8, 256, 0, stream>>>(xs, rnx, xrows);
    cos_rownorm_kernel<<<yrows / 8, 256, 0, stream>>>(spk, rny, yrows);

    dim3 grid(TDIM / 128, SDIM / 128, BATCH);  // (32, 2, 8)
    cos_wmma_gemm_kernel<<<grid, 256, 0, stream>>>(xs, spk, rnx, rny, out);
}